// MultiBoxLoss_87402584474174
// MI455X (gfx1250) — compile-verified
//
#include <hip/hip_runtime.h>
#include <math.h>

// ---------------- problem constants (fixed by reference setup) ----------------
#define PRIORS 8732
#define BATCH  32
#define NCLS   81
#define NOBJ   16
#define THRESHV 0.5f
#define NEGPOS  3
#define VAR0    0.1f
#define VAR1    0.2f

typedef float v2f __attribute__((ext_vector_type(2)));
typedef float v8f __attribute__((ext_vector_type(8)));

#if defined(__gfx1250__) && __has_builtin(__builtin_amdgcn_wmma_f32_16x16x4_f32)
#define HAVE_WMMA 1
#else
#define HAVE_WMMA 0
#endif

#if defined(__gfx1250__) && __has_builtin(__builtin_amdgcn_global_load_async_to_lds_b32)
#define HAVE_ASYNC 1
#else
#define HAVE_ASYNC 0
#endif

typedef __attribute__((address_space(1))) int* gptr_t;
typedef __attribute__((address_space(3))) int* lptr_t;

// =============================================================================
// Kernel 1: per-batch prior<->truth matching, loc encode, smooth-L1 partials
// =============================================================================
__global__ void __launch_bounds__(256)
match_kernel(const float* __restrict__ loc, const float* __restrict__ targets,
             const float* __restrict__ priors, int* __restrict__ conf_t,
             float* __restrict__ lossl_part, int* __restrict__ num_pos)
{
    const int b   = blockIdx.x;
    const int tid = threadIdx.x;
    const int lane = tid & 31;
    const int wid  = tid >> 5;

    __shared__ float tx1[NOBJ], ty1[NOBJ], tx2[NOBJ], ty2[NOBJ], tarea[NOBJ];
    __shared__ int   tlab[NOBJ];
    __shared__ float wov[8][NOBJ];
    __shared__ int   widx[8][NOBJ];
    __shared__ int   bpidx[NOBJ];
    __shared__ float red_f[256];
    __shared__ int   red_i[256];

    if (tid < NOBJ) {
        const float* tp = targets + ((size_t)b * NOBJ + tid) * 5;
        float x1 = tp[0], y1 = tp[1], x2 = tp[2], y2 = tp[3];
        tx1[tid] = x1; ty1[tid] = y1; tx2[tid] = x2; ty2[tid] = y2;
        tarea[tid] = (x2 - x1) * (y2 - y1);
        tlab[tid]  = (int)tp[4];
    }
    __syncthreads();

    // ---- pass 1: per-truth best prior (argmax over priors, first-max ties) ----
    float bo[NOBJ]; int bi[NOBJ];
#pragma unroll
    for (int t = 0; t < NOBJ; t++) { bo[t] = -1.0f; bi[t] = 0x7fffffff; }

    for (int p = tid; p < PRIORS; p += 256) {
        const float* pr = priors + (size_t)p * 4;
        float pcx = pr[0], pcy = pr[1], pw = pr[2], ph = pr[3];
        float px1 = pcx - pw * 0.5f, py1 = pcy - ph * 0.5f;
        float px2 = pcx + pw * 0.5f, py2 = pcy + ph * 0.5f;
        float pa = pw * ph;
#pragma unroll
        for (int t = 0; t < NOBJ; t++) {
            float ix1 = fmaxf(tx1[t], px1), iy1 = fmaxf(ty1[t], py1);
            float ix2 = fminf(tx2[t], px2), iy2 = fminf(ty2[t], py2);
            float iw = fmaxf(ix2 - ix1, 0.f), ih = fmaxf(iy2 - iy1, 0.f);
            float inter = iw * ih;
            float iou = inter / (tarea[t] + pa - inter);
            if (iou > bo[t]) { bo[t] = iou; bi[t] = p; }   // strict > keeps first max
        }
    }
    // wave reduce (prefer smaller prior index on ties, matching jnp.argmax)
#pragma unroll
    for (int off = 16; off > 0; off >>= 1) {
#pragma unroll
        for (int t = 0; t < NOBJ; t++) {
            float oov = __shfl_down(bo[t], off);
            int   oid = __shfl_down(bi[t], off);
            if (oov > bo[t] || (oov == bo[t] && oid < bi[t])) { bo[t] = oov; bi[t] = oid; }
        }
    }
    if (lane == 0) {
#pragma unroll
        for (int t = 0; t < NOBJ; t++) { wov[wid][t] = bo[t]; widx[wid][t] = bi[t]; }
    }
    __syncthreads();
    if (tid < NOBJ) {
        float best = -1.f; int bidx = 0x7fffffff;
        for (int w = 0; w < 8; w++) {
            float v = wov[w][tid]; int ix = widx[w][tid];
            if (v > best || (v == best && ix < bidx)) { best = v; bidx = ix; }
        }
        bpidx[tid] = bidx;
    }
    __syncthreads();

    // ---- pass 2: per-prior best truth + forced matches + encode + loss ----
    float lsum = 0.f; int pcnt = 0;
    for (int p = tid; p < PRIORS; p += 256) {
        const float* pr = priors + (size_t)p * 4;
        float pcx = pr[0], pcy = pr[1], pw = pr[2], ph = pr[3];
        float px1 = pcx - pw * 0.5f, py1 = pcy - ph * 0.5f;
        float px2 = pcx + pw * 0.5f, py2 = pcy + ph * 0.5f;
        float pa = pw * ph;
        float bov = -1.f; int bidx = 0;
#pragma unroll
        for (int t = 0; t < NOBJ; t++) {
            float ix1 = fmaxf(tx1[t], px1), iy1 = fmaxf(ty1[t], py1);
            float ix2 = fminf(tx2[t], px2), iy2 = fminf(ty2[t], py2);
            float iw = fmaxf(ix2 - ix1, 0.f), ih = fmaxf(iy2 - iy1, 0.f);
            float inter = iw * ih;
            float iou = inter / (tarea[t] + pa - inter);
            if (iou > bov) { bov = iou; bidx = t; }        // first max over truths
        }
#pragma unroll
        for (int t = 0; t < NOBJ; t++)                      // forced matches, last t wins
            if (bpidx[t] == p) { bov = 2.0f; bidx = t; }

        int cls = (bov < THRESHV) ? 0 : (tlab[bidx] + 1);
        conf_t[(size_t)b * PRIORS + p] = cls;

        if (cls > 0) {
            pcnt++;
            float mx1 = tx1[bidx], my1 = ty1[bidx], mx2 = tx2[bidx], my2 = ty2[bidx];
            float g0 = ((mx1 + mx2) * 0.5f - pcx) / (VAR0 * pw);
            float g1 = ((my1 + my2) * 0.5f - pcy) / (VAR0 * ph);
            float g2 = logf((mx2 - mx1) / pw) / VAR1;
            float g3 = logf((my2 - my1) / ph) / VAR1;
            const float* ld = loc + ((size_t)b * PRIORS + p) * 4;
            float g[4] = { g0, g1, g2, g3 };
#pragma unroll
            for (int c = 0; c < 4; c++) {
                float d = fabsf(ld[c] - g[c]);
                lsum += (d < 1.f) ? 0.5f * d * d : d - 0.5f;
            }
        }
    }

    red_f[tid] = lsum; red_i[tid] = pcnt;
    __syncthreads();
    for (int s = 128; s > 0; s >>= 1) {
        if (tid < s) { red_f[tid] += red_f[tid + s]; red_i[tid] += red_i[tid + s]; }
        __syncthreads();
    }
    if (tid == 0) { lossl_part[b] = red_f[0]; num_pos[b] = red_i[0]; }
}

// =============================================================================
// Kernel 2: cross-entropy per (b,p): async-stage conf tile to LDS, row max,
//           WMMA exp row-sum (B = ones), ce = mx + log(su) - picked
// =============================================================================
#define TILES_PER_B ((PRIORS + 15) / 16)   // 546

__global__ void __launch_bounds__(256)
ce_kernel(const float* __restrict__ conf, const int* __restrict__ conf_t,
          float* __restrict__ ce)
{
    __shared__ float sh[8 * 16 * NCLS];    // 8 waves x (16 priors x 81 classes)
    const int tid  = threadIdx.x;
    const int lane = tid & 31;
    const int wid  = tid >> 5;

    int tile = blockIdx.x * 8 + wid;
    if (tile >= BATCH * TILES_PER_B) return;   // uniform per wave
    int b  = tile / TILES_PER_B;
    int t  = tile - b * TILES_PER_B;
    int p0 = t * 16;
    int valid = min(16, PRIORS - p0);

    float* s = sh + wid * (16 * NCLS);
    const float* g = conf + ((size_t)b * PRIORS + p0) * NCLS;
    const int n = valid * NCLS;

#if HAVE_ASYNC
    for (int i = lane; i < 16 * NCLS; i += 32) {
        if (i < n) {
            __builtin_amdgcn_global_load_async_to_lds_b32(
                (gptr_t)(g + i), (lptr_t)(s + i), 0, 0);
        } else {
            s[i] = 0.f;
        }
    }
#if __has_builtin(__builtin_amdgcn_s_wait_asynccnt)
    __builtin_amdgcn_s_wait_asynccnt(0);
    asm volatile("" ::: "memory");
#else
    asm volatile("s_wait_asynccnt 0" ::: "memory");
#endif
#else
    for (int i = lane; i < 16 * NCLS; i += 32) s[i] = (i < n) ? g[i] : 0.f;
#endif

    // row max: lanes 0-15 scan cols [0,41) of row=lane, lanes 16-31 scan [41,81)
    const int m = lane & 15;
    float mx = -3.4e38f;
    const int c_lo = (lane < 16) ? 0 : 41;
    const int c_hi = (lane < 16) ? 41 : NCLS;
    for (int c = c_lo; c < c_hi; c++) mx = fmaxf(mx, s[m * NCLS + c]);
    mx = fmaxf(mx, __shfl_xor(mx, 16));

#if HAVE_WMMA
    // A(16x4 f32): lanes 0-15 hold K0,K1 in v0,v1; lanes 16-31 hold K2,K3.
    // B = all-ones -> every column of D is the row-sum of A (accumulated in C).
    v8f acc = {};
    v2f bones; bones.x = 1.f; bones.y = 1.f;
    const int kbase = (lane >> 4) << 1;
    for (int k0 = 0; k0 < NCLS; k0 += 4) {
        int c0 = k0 + kbase;
        float a0 = (c0     < NCLS) ? __expf(s[m * NCLS + c0]     - mx) : 0.f;
        float a1 = (c0 + 1 < NCLS) ? __expf(s[m * NCLS + c0 + 1] - mx) : 0.f;
        v2f a; a.x = a0; a.y = a1;
        acc = __builtin_amdgcn_wmma_f32_16x16x4_f32(
            false, a, false, bones, (short)0, acc, false, false);
    }
    // extract row sums: rows 0-7 live in lanes 0-15 (vgpr=row), rows 8-15 in lanes 16-31
    float su_own = 0.f;
#pragma unroll
    for (int r = 0; r < 8; r++) su_own = ((lane & 7) == r) ? acc[r] : su_own;
    float su_oth = __shfl_xor(su_own, 16);
    float su = ((lane & 8) == 0) ? su_own : su_oth;
#else
    float part = 0.f;
    const int kbase = (lane >> 4) << 1;
    for (int k0 = 0; k0 < NCLS; k0 += 4) {
        int c0 = k0 + kbase;
        if (c0     < NCLS) part += __expf(s[m * NCLS + c0]     - mx);
        if (c0 + 1 < NCLS) part += __expf(s[m * NCLS + c0 + 1] - mx);
    }
    float su = part + __shfl_xor(part, 16);
#endif

    if (lane < 16 && m < valid) {
        size_t idx = (size_t)b * PRIORS + p0 + m;
        int ct = conf_t[idx];
        float lse = mx + __logf(su);
        ce[idx] = lse - s[m * NCLS + ct];
    }
}

// =============================================================================
// Kernel 3: hard negative mining (exact stable rank) + pos/neg CE partials
// =============================================================================
__global__ void __launch_bounds__(256)
mine_kernel(const int* __restrict__ conf_t, const float* __restrict__ ce,
            const int* __restrict__ num_pos,
            float* __restrict__ posce_part, float* __restrict__ negce_part)
{
    const int b = blockIdx.x;
    const int tid = threadIdx.x;
    __shared__ float smine[PRIORS];   // 34928 B
    __shared__ float red[256];
    const size_t base = (size_t)b * PRIORS;

    float pos_sum = 0.f;
    for (int p = tid; p < PRIORS; p += 256) {
        int   ct = conf_t[base + p];
        float c  = ce[base + p];
        smine[p] = (ct > 0) ? 0.f : c;
        if (ct > 0) pos_sum += c;
    }
    __syncthreads();

    const int nneg = min(NEGPOS * num_pos[b], PRIORS - 1);
    float neg_sum = 0.f;
    const float4* s4 = (const float4*)smine;
    for (int p = tid; p < PRIORS; p += 256) {
        float mp = smine[p];
        int cnt = 0;
        for (int q4 = 0; q4 < PRIORS / 4; q4++) {
            float4 v = s4[q4];
            int q = q4 * 4;
            cnt += (v.x > mp) || (v.x == mp && (q + 0) < p);
            cnt += (v.y > mp) || (v.y == mp && (q + 1) < p);
            cnt += (v.z > mp) || (v.z == mp && (q + 2) < p);
            cnt += (v.w > mp) || (v.w == mp && (q + 3) < p);
        }
        if (cnt < nneg) neg_sum += ce[base + p];
    }

    red[tid] = pos_sum; __syncthreads();
    for (int s = 128; s > 0; s >>= 1) { if (tid < s) red[tid] += red[tid + s]; __syncthreads(); }
    if (tid == 0) posce_part[b] = red[0];
    __syncthreads();
    red[tid] = neg_sum; __syncthreads();
    for (int s = 128; s > 0; s >>= 1) { if (tid < s) red[tid] += red[tid + s]; __syncthreads(); }
    if (tid == 0) negce_part[b] = red[0];
}

// =============================================================================
// Kernel 4: deterministic final reduction -> (loss_l/N, loss_c/N)
// =============================================================================
__global__ void finalize_kernel(const float* __restrict__ lossl_part,
                                const float* __restrict__ posce_part,
                                const float* __restrict__ negce_part,
                                const int* __restrict__ num_pos,
                                float* __restrict__ out)
{
    if (threadIdx.x == 0 && blockIdx.x == 0) {
        float ll = 0.f, lc = 0.f; int np = 0;
        for (int b = 0; b < BATCH; b++) {
            ll += lossl_part[b];
            lc += posce_part[b] + negce_part[b];
            np += num_pos[b];
        }
        float N = (float)np;
        out[0] = ll / N;   // LOC_WEIGHT == 1.0
        out[1] = lc / N;
    }
}

// =============================================================================
// launch: ws layout (bytes): conf_t[B*P] int | ce[B*P] f32 | lossl[32] |
//         num_pos[32] int | posce[32] | negce[32]   (~2.24 MB total)
// =============================================================================
extern "C" void kernel_launch(void* const* d_in, const int* in_sizes, int n_in,
                              void* d_out, int out_size, void* d_ws, size_t ws_size,
                              hipStream_t stream)
{
    const float* loc     = (const float*)d_in[0];
    const float* conf    = (const float*)d_in[1];
    const float* targets = (const float*)d_in[2];
    const float* priors  = (const float*)d_in[3];

    const size_t BP = (size_t)BATCH * PRIORS;
    int*   conf_t     = (int*)d_ws;
    float* ce         = (float*)d_ws + BP;
    float* lossl_part = (float*)d_ws + 2 * BP;
    int*   num_pos    = (int*)d_ws + 2 * BP + 32;
    float* posce      = (float*)d_ws + 2 * BP + 64;
    float* negce      = (float*)d_ws + 2 * BP + 96;
    float* out        = (float*)d_out;

    hipLaunchKernelGGL(match_kernel, dim3(BATCH), dim3(256), 0, stream,
                       loc, targets, priors, conf_t, lossl_part, num_pos);

    const int total_tiles = BATCH * TILES_PER_B;          // 17472
    const int blocks = (total_tiles + 7) / 8;             // 2184
    hipLaunchKernelGGL(ce_kernel, dim3(blocks), dim3(256), 0, stream,
                       conf, conf_t, ce);

    hipLaunchKernelGGL(mine_kernel, dim3(BATCH), dim3(256), 0, stream,
                       conf_t, ce, num_pos, posce, negce);

    hipLaunchKernelGGL(finalize_kernel, dim3(1), dim3(32), 0, stream,
                       lossl_part, posce, negce, num_pos, out);
}